// Noise_Regressor_67774583930935
// MI455X (gfx1250) — compile-verified
//
#include <hip/hip_runtime.h>
#include <math.h>

typedef float v2f __attribute__((ext_vector_type(2)));
typedef float v8f __attribute__((ext_vector_type(8)));

#define SEQ    2048
#define DMODEL 1024
#define NIMU   13
#define NOUT   130           // 10 params * 13 imus
#define NPAD   144           // 130 padded to 9*16 for WMMA tiles
#define LN_EPS 1e-5f

// workspace layout (in floats)
#define OFF_XN   0                           // 2048*1024      = 8 MB
#define OFF_WP   (SEQ*DMODEL)                // 1024*144 padded W
#define OFF_P    (OFF_WP + DMODEL*NPAD)      // 2048*144 projection out
#define OFF_PAR  (OFF_P + SEQ*NPAD)          // 8 arrays * 13*2048 (c,hd,om,ph, cth,hdt,omt,pht)
#define OFF_PART (OFF_PAR + 8*NIMU*SEQ)      // 8 i-chunks * 13*2048 partial sums

__device__ __forceinline__ float softplus_f(float x) {
    // numerically stable softplus: max(x,0) + log(1+exp(-|x|))
    return fmaxf(x, 0.0f) + __logf(1.0f + __expf(-fabsf(x)));
}

// ---------------- pad W (1024x130) into (1024x144) with zero fill ----------------
__global__ void pad_w_kernel(const float* __restrict__ W, float* __restrict__ wp) {
    int idx = blockIdx.x * 256 + threadIdx.x;          // < 1024*144
    int k = idx / NPAD, n = idx % NPAD;
    wp[idx] = (n < NOUT) ? W[k * NOUT + n] : 0.0f;
}

// ---------------- LayerNorm: one block per row ----------------
__global__ void __launch_bounds__(256) ln_kernel(const float* __restrict__ x,
                                                 const float* __restrict__ gamma,
                                                 const float* __restrict__ beta,
                                                 float* __restrict__ xn) {
    __shared__ float s1[256];
    __shared__ float s2[256];
    const int row = blockIdx.x;
    const float* xr = x + row * DMODEL;
    float v[4], sum = 0.0f, ss = 0.0f;
#pragma unroll
    for (int q = 0; q < 4; q++) {
        v[q] = xr[threadIdx.x + 256 * q];
        sum += v[q];
        ss  += v[q] * v[q];
    }
    s1[threadIdx.x] = sum;
    s2[threadIdx.x] = ss;
    __syncthreads();
    for (int st = 128; st > 0; st >>= 1) {
        if (threadIdx.x < st) {
            s1[threadIdx.x] += s1[threadIdx.x + st];
            s2[threadIdx.x] += s2[threadIdx.x + st];
        }
        __syncthreads();
    }
    const float mu  = s1[0] * (1.0f / DMODEL);
    const float var = s2[0] * (1.0f / DMODEL) - mu * mu;
    const float inv = rsqrtf(var + LN_EPS);
#pragma unroll
    for (int q = 0; q < 4; q++) {
        int d = threadIdx.x + 256 * q;
        xn[row * DMODEL + d] = (v[q] - mu) * inv * gamma[d] + beta[d];
    }
}

// ---------------- GEMM via native f32 WMMA: one wave per 16x16 tile ----------------
__global__ void __launch_bounds__(32) gemm_wmma_kernel(const float* __restrict__ xn,
                                                       const float* __restrict__ wp,
                                                       const float* __restrict__ bias,
                                                       float* __restrict__ P) {
    const int m0   = blockIdx.x * 16;
    const int n0   = blockIdx.y * 16;
    const int lane = threadIdx.x;                // 0..31, wave32
    const int mrow = lane & 15;                  // A: M = lane%16
    const int kh   = (lane >> 4) * 2;            // A/B: lanes 0-15 hold K={0,1}, 16-31 hold K={2,3}
    const int ncol = lane & 15;                  // B/C/D: N = lane%16

    const float* xr = xn + (m0 + mrow) * DMODEL + kh;
    const float* wc = wp + n0 + ncol;

    v8f acc = {};
    for (int k0 = 0; k0 < DMODEL; k0 += 4) {
        v2f a, b;
        a.x = xr[k0];
        a.y = xr[k0 + 1];
        b.x = wc[(k0 + kh) * NPAD];
        b.y = wc[(k0 + kh + 1) * NPAD];
        // V_WMMA_F32_16X16X4_F32 : D = A(16x4) * B(4x16) + C, all f32
        acc = __builtin_amdgcn_wmma_f32_16x16x4_f32(
            /*neg_a=*/false, a, /*neg_b=*/false, b,
            /*c_mod=*/(short)0, acc, /*reuse_a=*/false, /*reuse_b=*/false);
    }

    const int col = n0 + ncol;
    const float bv = (col < NOUT) ? bias[col] : 0.0f;
    const int mbase = m0 + ((lane >> 4) * 8);    // D: VGPR r -> M=r (lanes<16) / r+8 (lanes>=16)
#pragma unroll
    for (int r = 0; r < 8; r++) {
        P[(mbase + r) * NPAD + col] = acc[r] + bv;
    }
}

// ---------------- param transforms: SoA arrays for the kinematic kernel ----------------
__global__ void __launch_bounds__(256) params_kernel(const float* __restrict__ P,
                                                     float* __restrict__ par,
                                                     float* __restrict__ nstd_out) {
    const int idx = blockIdx.x * 256 + threadIdx.x;  // < 13*2048  (imu-major: imu*SEQ + i)
    const int imu = idx >> 11;
    const int i   = idx & 2047;
    const float* pr = P + i * NPAD;

    const float p0  = pr[0 * NIMU + imu];
    const float p1  = pr[1 * NIMU + imu];
    const float p2  = pr[2 * NIMU + imu];
    const float p3  = pr[3 * NIMU + imu];
    const float c   = pr[4 * NIMU + imu];
    const float cth = pr[5 * NIMU + imu];
    const float ph  = pr[6 * NIMU + imu];
    const float pht = pr[7 * NIMU + imu];
    const float p9  = pr[9 * NIMU + imu];

    const float d   = softplus_f(p1);
    const float dth = softplus_f(p3);
    // omega = sqrt(4k - d^2)/2 with k = d^2/4 + softplus(p0)  ==  sqrt(softplus(p0))
    const int stride = NIMU * SEQ;
    par[0 * stride + idx] = c;
    par[1 * stride + idx] = 0.5f * d;
    par[2 * stride + idx] = sqrtf(softplus_f(p0));
    par[3 * stride + idx] = ph;
    par[4 * stride + idx] = cth;
    par[5 * stride + idx] = 0.5f * dth;
    par[6 * stride + idx] = sqrtf(softplus_f(p2));
    par[7 * stride + idx] = pht;

    nstd_out[idx] = softplus_f(p9);              // noise_std[imu, j] laid out imu*SEQ + j
}

// ---------------- causal damped-sinusoid accumulation ----------------
// grid: (SEQ/64 j-tiles, 13 imus, 8 i-chunks of 256). Block = 64 threads, one j each.
__global__ void __launch_bounds__(64) kin_kernel(const float* __restrict__ par,
                                                 float* __restrict__ part) {
    __shared__ float lds[8 * 256];               // 8 param arrays x 256-i chunk = 8KB
    const int j    = blockIdx.x * 64 + (int)threadIdx.x;
    const int imu  = blockIdx.y;
    const int ic   = blockIdx.z;
    const int ibase = ic * 256;
    const int jmax = blockIdx.x * 64 + 63;

    float acc = 0.0f;
    if (ibase <= jmax) {                         // block-uniform condition
        const int stride = NIMU * SEQ;
        for (int t = threadIdx.x; t < 8 * 256; t += 64) {
            int a = t >> 8, e = t & 255;
            lds[t] = par[a * stride + imu * SEQ + ibase + e];
        }
        __syncthreads();
        int emax = jmax - ibase;
        if (emax > 255) emax = 255;
        for (int e = 0; e <= emax; e++) {
            const int i = ibase + e;
            if (i <= j) {
                const float tt  = (float)(j - i);
                const float lin = lds[e] * __expf(-lds[256 + e] * tt)
                                * __sinf(fmaf(lds[512 + e], tt, lds[768 + e]));
                const float ang = lds[1024 + e] * __expf(-lds[1280 + e] * tt)
                                * __sinf(fmaf(lds[1536 + e], tt, lds[1792 + e]));
                acc += lin + ang;
            }
        }
    }
    part[(ic * NIMU + imu) * SEQ + j] = acc;     // always written (no stale ws data)
}

// ---------------- final reduce over i-chunks + bias + scalar ----------------
__global__ void __launch_bounds__(256) reduce_kernel(const float* __restrict__ part,
                                                     const float* __restrict__ P,
                                                     const float* __restrict__ mo,
                                                     float* __restrict__ out) {
    const int idx = blockIdx.x * 256 + threadIdx.x;  // < 13*2048
    const int imu = idx >> 11;
    const int j   = idx & 2047;
    float s = 0.0f;
#pragma unroll
    for (int ic = 0; ic < 8; ic++) s += part[(ic * NIMU + imu) * SEQ + j];
    out[idx] = s + mo[0] + P[j * NPAD + 8 * NIMU + imu];   // + min_orig + noise_bias
}

extern "C" void kernel_launch(void* const* d_in, const int* in_sizes, int n_in,
                              void* d_out, int out_size, void* d_ws, size_t ws_size,
                              hipStream_t stream) {
    const float* hs   = (const float*)d_in[0];   // (1, 2048, 1024)
    const float* mo   = (const float*)d_in[1];   // scalar
    const float* g    = (const float*)d_in[2];   // (1024,)
    const float* be   = (const float*)d_in[3];   // (1024,)
    const float* W    = (const float*)d_in[4];   // (1024, 130)
    const float* bias = (const float*)d_in[5];   // (130,)

    float* ws   = (float*)d_ws;
    float* XN   = ws + OFF_XN;
    float* WP   = ws + OFF_WP;
    float* P    = ws + OFF_P;
    float* PAR  = ws + OFF_PAR;
    float* PART = ws + OFF_PART;
    float* out  = (float*)d_out;

    pad_w_kernel<<<(DMODEL * NPAD) / 256, 256, 0, stream>>>(W, WP);
    ln_kernel<<<SEQ, 256, 0, stream>>>(hs, g, be, XN);
    gemm_wmma_kernel<<<dim3(SEQ / 16, NPAD / 16), 32, 0, stream>>>(XN, WP, bias, P);
    params_kernel<<<(NIMU * SEQ) / 256, 256, 0, stream>>>(P, PAR, out + NIMU * SEQ);
    kin_kernel<<<dim3(SEQ / 64, NIMU, 8), 64, 0, stream>>>(PAR, PART);
    reduce_kernel<<<(NIMU * SEQ) / 256, 256, 0, stream>>>(PART, P, mo, out);
}